// PosAttention_41326175322451
// MI455X (gfx1250) — compile-verified
//
#include <hip/hip_runtime.h>

// CDNA5 (gfx1250) wave32 WMMA types
typedef float v2f __attribute__((ext_vector_type(2)));
typedef float v8f __attribute__((ext_vector_type(8)));

// D = A(16x4 f32) * B(4x16 f32) + C(16x16 f32)
// args: (neg_a, A, neg_b, B, c_mod, C, reuse_a, reuse_b)
#define WMMA_F32(a, b, c) \
  __builtin_amdgcn_wmma_f32_16x16x4_f32(false, (a), false, (b), (short)0, (c), false, false)

constexpr int kB = 8, kN = 1024, kP = 128, kH = 64, kD = 64, kJ = 256, kX = 64;
constexpr int INNER = kH * kD;   // 4096
constexpr int CH = 32;           // token chunk
constexpr int NCH = kN / CH;     // 32 chunks

// ---------------------------------------------------------------------------
// Kernel A: fused q-projection + latent attention (flash-style online softmax)
// grid = 512 (b*64 + h), block = 256 (8 waves). Each wave owns 32 latents.
// Writes normalized attention output ws[b][j][h][xd] (f32, 32 MB).
// ---------------------------------------------------------------------------
__global__ __launch_bounds__(256)
void pos_attn_fused(const float* __restrict__ pos, const float* __restrict__ x,
                    const float* __restrict__ Wq, const float* __restrict__ kl,
                    float* __restrict__ ws) {
  __shared__ float lds_x[CH][kX];        // 8 KB   x chunk   [n][xd]
  __shared__ float lds_qh[CH][kD + 1];   // 8.3 KB Qh chunk  [n][d] (padded)
  __shared__ float lds_p[8][32][CH + 1]; // 33.8KB per-wave P [j][n] (padded)

  const int bb   = blockIdx.x >> 6;
  const int h    = blockIdx.x & 63;
  const int tid  = threadIdx.x;
  const int w    = tid >> 5;        // wave id 0..7
  const int lane = tid & 31;
  const int row  = lane & 15;       // A-frag row / B-frag col / C-frag col
  const int half = lane >> 4;
  const int kb   = half * 2;        // K sub-offset inside a 16x4 / 4x16 frag

  const int tt = w >> 2;            // token tile for phase 2 (0..1)
  const int dt = w & 3;             // d tile for phase 2 (0..3)

  v8f   oacc[2][4];
  float mrun[2][8], lrun[2][8];
#pragma unroll
  for (int jt = 0; jt < 2; ++jt) {
#pragma unroll
    for (int xt = 0; xt < 4; ++xt) oacc[jt][xt] = (v8f){0,0,0,0,0,0,0,0};
#pragma unroll
    for (int r = 0; r < 8; ++r) { mrun[jt][r] = -3.0e38f; lrun[jt][r] = 0.0f; }
  }

  const float* posB = pos + (size_t)bb * kN * kP;
  const float* xB   = x   + (size_t)bb * kN * kX;

  for (int ch = 0; ch < NCH; ++ch) {
    const int n0 = ch * CH;
    __syncthreads();  // protect lds_x / lds_qh from previous iteration readers

    // ---- stage x chunk: 32x64 f32, contiguous, float4 cooperative load ----
    {
      const float4* gx = reinterpret_cast<const float4*>(xB + n0 * kX);
      float4* sx = reinterpret_cast<float4*>(&lds_x[0][0]);
#pragma unroll
      for (int i = 0; i < 2; ++i) sx[tid + i * 256] = gx[tid + i * 256];
    }

    // ---- phase 2: Qh = pos_chunk @ Wq[:, h*64 : h*64+64], one tile/wave ----
    {
      v8f cq = (v8f){0,0,0,0,0,0,0,0};
      const float* pA = posB + (size_t)(n0 + tt * 16 + row) * kP;
      const float* pB = Wq + h * kD + dt * 16 + row;  // row acts as col here
#pragma unroll
      for (int k = 0; k < kP; k += 4) {
        v2f a = *reinterpret_cast<const v2f*>(pA + k + kb);
        v2f b;
        b.x = pB[(size_t)(k + kb) * INNER];
        b.y = pB[(size_t)(k + kb + 1) * INNER];
        cq = WMMA_F32(a, b, cq);
      }
#pragma unroll
      for (int r = 0; r < 8; ++r)
        lds_qh[tt * 16 + r + 8 * half][dt * 16 + row] = cq[r];
    }
    __syncthreads();  // Qh + x visible to all waves

    // ---- phase 3: S = k_latent(32 rows/wave) @ Qh^T, scaled ----
    v8f s[2][2];
#pragma unroll
    for (int jt = 0; jt < 2; ++jt) {
#pragma unroll
      for (int nt = 0; nt < 2; ++nt) {
        v8f c = (v8f){0,0,0,0,0,0,0,0};
        const float* aK = kl + (size_t)(w * 32 + jt * 16 + row) * kD;
#pragma unroll
        for (int k = 0; k < kD; k += 4) {
          v2f a = *reinterpret_cast<const v2f*>(aK + k + kb);
          v2f b;
          b.x = lds_qh[nt * 16 + row][k + kb];      // B[kk][n] = Qh[n][kk]
          b.y = lds_qh[nt * 16 + row][k + kb + 1];
          c = WMMA_F32(a, b, c);
        }
#pragma unroll
        for (int r = 0; r < 8; ++r) c[r] *= 0.125f;  // d^-0.5
        s[jt][nt] = c;
      }
    }

    // ---- phase 4: online softmax over tokens (rows live in C layout) ----
#pragma unroll
    for (int jt = 0; jt < 2; ++jt) {
      float t[8], mnew[8], fsc[8], rs[8];
#pragma unroll
      for (int r = 0; r < 8; ++r) t[r] = fmaxf(s[jt][0][r], s[jt][1][r]);
#pragma unroll
      for (int off = 1; off <= 8; off <<= 1)
#pragma unroll
        for (int r = 0; r < 8; ++r)
          t[r] = fmaxf(t[r], __shfl_xor(t[r], off, 32));  // within 16-lane half
#pragma unroll
      for (int r = 0; r < 8; ++r) {
        mnew[r] = fmaxf(mrun[jt][r], t[r]);
        fsc[r]  = __expf(mrun[jt][r] - mnew[r]);
      }
#pragma unroll
      for (int nt = 0; nt < 2; ++nt)
#pragma unroll
        for (int r = 0; r < 8; ++r)
          s[jt][nt][r] = __expf(s[jt][nt][r] - mnew[r]);  // P
#pragma unroll
      for (int r = 0; r < 8; ++r) rs[r] = s[jt][0][r] + s[jt][1][r];
#pragma unroll
      for (int off = 1; off <= 8; off <<= 1)
#pragma unroll
        for (int r = 0; r < 8; ++r) rs[r] += __shfl_xor(rs[r], off, 32);
#pragma unroll
      for (int r = 0; r < 8; ++r) {
        lrun[jt][r] = lrun[jt][r] * fsc[r] + rs[r];
        mrun[jt][r] = mnew[r];
      }
#pragma unroll
      for (int xt = 0; xt < 4; ++xt)
#pragma unroll
        for (int r = 0; r < 8; ++r) oacc[jt][xt][r] *= fsc[r];
      // stage P through per-wave LDS: C layout -> A layout transpose
#pragma unroll
      for (int nt = 0; nt < 2; ++nt)
#pragma unroll
        for (int r = 0; r < 8; ++r)
          lds_p[w][jt * 16 + r + 8 * half][nt * 16 + row] = s[jt][nt][r];
    }

    // ---- phase 5: O += P @ x_chunk (per-wave LDS, no barrier needed) ----
#pragma unroll
    for (int jt = 0; jt < 2; ++jt) {
#pragma unroll
      for (int xt = 0; xt < 4; ++xt) {
        v8f c = oacc[jt][xt];
#pragma unroll
        for (int k = 0; k < CH; k += 4) {
          v2f a, b;
          a.x = lds_p[w][jt * 16 + row][k + kb];
          a.y = lds_p[w][jt * 16 + row][k + kb + 1];
          b.x = lds_x[k + kb][xt * 16 + row];
          b.y = lds_x[k + kb + 1][xt * 16 + row];
          c = WMMA_F32(a, b, c);
        }
        oacc[jt][xt] = c;
      }
    }
  }

  // ---- epilogue: normalize by 1/l, store ws[b][j][h][xd] ----
#pragma unroll
  for (int jt = 0; jt < 2; ++jt) {
    float inv[8];
#pragma unroll
    for (int r = 0; r < 8; ++r) inv[r] = 1.0f / lrun[jt][r];
#pragma unroll
    for (int xt = 0; xt < 4; ++xt)
#pragma unroll
      for (int r = 0; r < 8; ++r) {
        const int j  = w * 32 + jt * 16 + r + 8 * half;
        const int xd = xt * 16 + row;
        ws[(size_t)(bb * kJ + j) * INNER + h * kX + xd] = oacc[jt][xt][r] * inv[r];
      }
  }
}

// ---------------------------------------------------------------------------
// Kernel B: out = ws(2048 x 4096) @ W_out(4096 x 64) + b_out
// grid = 128 row-tiles, block = 128 (4 waves = 4 col-tiles of 16)
// ---------------------------------------------------------------------------
__global__ __launch_bounds__(128)
void out_proj(const float* __restrict__ ws, const float* __restrict__ Wout,
              const float* __restrict__ bout, float* __restrict__ out) {
  const int rt   = blockIdx.x;
  const int w    = threadIdx.x >> 5;
  const int lane = threadIdx.x & 31;
  const int row  = lane & 15;
  const int half = lane >> 4;
  const int kb   = half * 2;

  const float* aR = ws + (size_t)(rt * 16 + row) * INNER;
  const float* bC = Wout + w * 16 + row;

  v8f c = (v8f){0,0,0,0,0,0,0,0};
  for (int k = 0; k < INNER; k += 4) {
    v2f a = *reinterpret_cast<const v2f*>(aR + k + kb);
    v2f b;
    b.x = bC[(size_t)(k + kb) * 64];
    b.y = bC[(size_t)(k + kb + 1) * 64];
    c = WMMA_F32(a, b, c);
  }
  const float bias = bout[w * 16 + row];
#pragma unroll
  for (int r = 0; r < 8; ++r)
    out[(size_t)(rt * 16 + r + 8 * half) * 64 + w * 16 + row] = c[r] + bias;
}

// ---------------------------------------------------------------------------
extern "C" void kernel_launch(void* const* d_in, const int* in_sizes, int n_in,
                              void* d_out, int out_size, void* d_ws, size_t ws_size,
                              hipStream_t stream) {
  (void)in_sizes; (void)n_in; (void)out_size; (void)ws_size;
  const float* pos  = (const float*)d_in[0];
  const float* x    = (const float*)d_in[1];
  const float* Wq   = (const float*)d_in[2];
  const float* kl   = (const float*)d_in[3];
  const float* Wout = (const float*)d_in[4];
  const float* bout = (const float*)d_in[5];
  float* ws  = (float*)d_ws;   // needs 8*256*4096*4 = 32 MB
  float* out = (float*)d_out;

  pos_attn_fused<<<kB * kH, 256, 0, stream>>>(pos, x, Wq, kl, ws);
  out_proj<<<(kB * kJ) / 16, 128, 0, stream>>>(ws, Wout, bout, out);
}